// GINEStack_48155173323267
// MI455X (gfx1250) — compile-verified
//
#include <hip/hip_runtime.h>
#include <hip/hip_bf16.h>

// ---------------------------------------------------------------------------
// GINE stack for MI455X (gfx1250, wave32).
//  * All GEMMs via V_WMMA_F32_16X16X4_F32 (full fp32 precision; workload is
//    gather/scatter bound so low-precision WMMA buys nothing).
//  * GEMM templated on (N, K): constant addressing, fully unrolled k-loop.
//  * A (80 rows, stride-65 padded) and W (64x64 chunk) both staged in LDS;
//    inner loop is pure DS + WMMA: 6 clause-able ds_loads -> one wait ->
//    5 back-to-back v_wmma (one B fragment feeds 5 stacked 16x16 tiles).
//  * h_l lives directly in d_out[:, l*128:(l+1)*128] (stride 512) -> free concat.
//  * Edge linear fused into the scatter kernel (We in LDS, f32 atomics to agg).
// ---------------------------------------------------------------------------

#define N_NODES 50000
#define N_EDGES 800000
#define NODE_F  128
#define EDGE_F  64
#define HID     128
#define N_LAYERS 3
#define OUT_C   ((N_LAYERS + 1) * HID)   // 512

#define MT 80        // M rows per block (50000 = 625 * 80, exact)
#define MSUB 5       // 5 x 16-row WMMA subtiles per wave
#define KT 64        // K chunk staged in LDS
#define APAD 65      // padded LDS row stride (conflict-free vs 64 banks)

typedef float v2f __attribute__((ext_vector_type(2)));
typedef float v8f __attribute__((ext_vector_type(8)));

__device__ __forceinline__ float gelu_exact(float x) {
    return 0.5f * x * (1.0f + erff(x * 0.70710678118654752440f));
}

// ---------------------------------------------------------------------------
// C[M x N] = A[M x K] @ W[K x N] + bias
// grid = (M/80, N/64), block = 128 threads = 4 waves; wave w owns columns
// [64*by + 16*w, +16) for all 80 rows of the block.
// VGPR layouts per CDNA5 ISA 7.12.2:
//   A 16x4 f32 : lanes 0-15 -> K = k0+0 (v0), k0+1 (v1); lanes 16-31 -> k0+2,k0+3
//   B 4x16 f32 : mirror of A (N across lanes, K split across lane halves)
//   C/D 16x16  : VGPR v, lanes 0-15 -> M=v, N=lane; lanes 16-31 -> M=v+8
// ---------------------------------------------------------------------------
template <int N, int K>
__global__ __launch_bounds__(128)
void gemm_wmma_f32(const float* __restrict__ A, int lda,
                   const float* __restrict__ W,      // K x N row-major
                   const float* __restrict__ bias,   // [N]
                   float* __restrict__ C, int ldc) {
    __shared__ float As[MT * APAD];                  // 80 x 64 chunk, padded
    __shared__ float Ws[KT * 64];                    // 64 x 64 chunk of W
    const int tid  = threadIdx.x;
    const int lane = tid & 31;
    const int wave = tid >> 5;
    const int rowBase  = blockIdx.x * MT;
    const int colBlock = blockIdx.y * 64;

    const int nlo  = lane & 15;          // N (for B/C) and M (for A) within tile
    const int koff = (lane >> 4) * 2;    // lane-half K split
    const int wcol = wave * 16 + nlo;    // column within the block's 64-wide strip
    const int col  = colBlock + wcol;

    v8f acc[MSUB];
    {
        float bv = bias[col];
        #pragma unroll
        for (int s = 0; s < MSUB; ++s)
            #pragma unroll
            for (int i = 0; i < 8; ++i) acc[s][i] = bv;
    }

    #pragma unroll 1
    for (int kb = 0; kb < K; kb += KT) {
        // cooperative load: A chunk (80 x 64, padded) and W chunk (64 x 64)
        #pragma unroll
        for (int i = 0; i < (MT * KT) / 128; ++i) {
            int idx = tid + i * 128;
            int r = idx >> 6, c = idx & 63;          // KT == 64
            As[r * APAD + c] = A[(size_t)(rowBase + r) * lda + kb + c];
        }
        #pragma unroll
        for (int i = 0; i < (KT * 64) / 128; ++i) {
            int idx = tid + i * 128;
            int r = idx >> 6, c = idx & 63;
            Ws[idx] = W[(size_t)(kb + r) * N + colBlock + c];
        }
        __syncthreads();

        #pragma unroll
        for (int k0 = 0; k0 < KT; k0 += 4) {
            v2f a[MSUB];
            #pragma unroll
            for (int s = 0; s < MSUB; ++s) {
                const float* arow = &As[(s * 16 + nlo) * APAD + k0 + koff];
                a[s].x = arow[0];
                a[s].y = arow[1];
            }
            v2f b;
            b.x = Ws[(k0 + koff)     * 64 + wcol];
            b.y = Ws[(k0 + koff + 1) * 64 + wcol];
            #pragma unroll
            for (int s = 0; s < MSUB; ++s)
                acc[s] = __builtin_amdgcn_wmma_f32_16x16x4_f32(
                    false, a[s], false, b, (short)0, acc[s], false, false);
        }
        __syncthreads();
    }

    const int rsel = (lane >> 4) * 8;
    #pragma unroll
    for (int s = 0; s < MSUB; ++s) {
        #pragma unroll
        for (int v = 0; v < 8; ++v) {
            int r = rowBase + s * 16 + v + rsel;
            C[(size_t)r * ldc + col] = acc[s][v];
        }
    }
}

// ---------------------------------------------------------------------------
// BatchNorm statistics: per-column sum / sum-of-squares over M rows.
// Block covers 256 rows; C in {128, 256}. Partial sums reduced in LDS, then
// one atomicAdd per column per block.
// ---------------------------------------------------------------------------
__global__ __launch_bounds__(256)
void bn_stats(const float* __restrict__ X, int M, int C,
              float* __restrict__ gsum, float* __restrict__ gsumsq) {
    __shared__ float ls[256], ls2[256];
    int tid  = threadIdx.x;
    int col  = tid & (C - 1);
    int step = 256 / C;
    int r    = blockIdx.x * 256 + tid / C;
    int rend = blockIdx.x * 256 + 256;
    if (rend > M) rend = M;
    float s = 0.f, s2 = 0.f;
    for (; r < rend; r += step) {
        float v = X[(size_t)r * C + col];
        s += v; s2 += v * v;
    }
    ls[tid] = s; ls2[tid] = s2;
    __syncthreads();
    if (C == 128 && tid < 128) { ls[tid] += ls[tid + 128]; ls2[tid] += ls2[tid + 128]; }
    __syncthreads();
    if (tid < C) {
        atomicAdd(&gsum[tid],   ls[tid]);
        atomicAdd(&gsumsq[tid], ls2[tid]);
    }
}

__global__ void bn_finalize(const float* __restrict__ gsum,
                            const float* __restrict__ gsumsq,
                            const float* __restrict__ g,
                            const float* __restrict__ beta,
                            int M,
                            float* __restrict__ scale, float* __restrict__ shift) {
    int f = threadIdx.x;
    float inv_m = 1.0f / (float)M;
    float mean  = gsum[f] * inv_m;
    float var   = gsumsq[f] * inv_m - mean * mean;   // biased var
    float rs    = rsqrtf(var + 1e-5f);
    float sc    = g[f] * rs;
    scale[f] = sc;
    shift[f] = beta[f] - sc * mean;
}

__global__ __launch_bounds__(256)
void bn_apply_gelu(const float* __restrict__ X, int ldin,
                   float* __restrict__ Y, int ldout,
                   const float* __restrict__ scale,
                   const float* __restrict__ shift,
                   int M, int C) {
    int gid = blockIdx.x * 256 + threadIdx.x;
    if (gid >= M * C) return;
    int row = gid / C, col = gid - row * C;
    float v = scale[col] * X[(size_t)row * ldin + col] + shift[col];
    Y[(size_t)row * ldout + col] = gelu_exact(v);
}

// ---------------------------------------------------------------------------
// Fused GINE message + scatter: for each edge e,
//   msg = relu(h[src] + edge_attr[e] @ We + be);  agg[dst] += msg (f32 atomics)
// One wave per edge; We (64x128 = 32KB) + be in LDS; attr row staged per-wave
// in LDS via b64 stores (wave-synchronous, DS in-order per wave). h read from
// the d_out slice (row stride 512); h table is L2-resident (25.6 MB << 192 MB).
// Next edge's attr row prefetched (global_prefetch_b8) to cover the 205 MB
// streaming edge_attr read.
// ---------------------------------------------------------------------------
__global__ __launch_bounds__(256)
void gine_edge_scatter(const float* __restrict__ edge_attr,   // [E, 64]
                       const int*   __restrict__ edge_index,  // [2, E]
                       const float* __restrict__ We,          // [64, 128]
                       const float* __restrict__ be,          // [128]
                       const float* __restrict__ hbase, int ldh,
                       float* __restrict__ agg) {
    __shared__ float WeS[EDGE_F * HID];
    __shared__ float beS[HID];
    __shared__ float attrS[8 * EDGE_F];
    int tid = threadIdx.x;
    for (int i = tid; i < EDGE_F * HID; i += 256) WeS[i] = We[i];
    if (tid < HID) beS[tid] = be[tid];
    __syncthreads();

    int lane = tid & 31, wave = tid >> 5;
    float2* aS2 = (float2*)&attrS[wave * EDGE_F];
    const float* aS = &attrS[wave * EDGE_F];
    int e  = blockIdx.x * 8 + wave;
    int nW = gridDim.x * 8;
    for (; e < N_EDGES; e += nW) {
        // stage this edge's 64 attrs (one b64 per lane); prefetch next edge's row
        aS2[lane] = ((const float2*)(edge_attr + (size_t)e * EDGE_F))[lane];
        if (e + nW < N_EDGES)
            __builtin_prefetch(edge_attr + (size_t)(e + nW) * EDGE_F + 2 * lane, 0, 1);

        int src = edge_index[e];
        int dst = edge_index[N_EDGES + e];

        float a0 = beS[lane], a1 = beS[lane + 32];
        float a2 = beS[lane + 64], a3 = beS[lane + 96];
        #pragma unroll 8
        for (int k = 0; k < EDGE_F; ++k) {
            float av = aS[k];
            const float* wr = &WeS[k * HID];
            a0 += av * wr[lane];
            a1 += av * wr[lane + 32];
            a2 += av * wr[lane + 64];
            a3 += av * wr[lane + 96];
        }
        const float* hs = &hbase[(size_t)src * ldh];
        float m0 = fmaxf(hs[lane]      + a0, 0.f);
        float m1 = fmaxf(hs[lane + 32] + a1, 0.f);
        float m2 = fmaxf(hs[lane + 64] + a2, 0.f);
        float m3 = fmaxf(hs[lane + 96] + a3, 0.f);
        float* ad = &agg[(size_t)dst * HID];
        atomicAdd(&ad[lane],      m0);
        atomicAdd(&ad[lane + 32], m1);
        atomicAdd(&ad[lane + 64], m2);
        atomicAdd(&ad[lane + 96], m3);
    }
}

// z0 = (1 + eps[l]) * h + agg  (in place on agg)
__global__ __launch_bounds__(256)
void z0_combine(float* __restrict__ agg,
                const float* __restrict__ hbase, int ldh,
                const float* __restrict__ eps, int layer) {
    int gid = blockIdx.x * 256 + threadIdx.x;
    if (gid >= N_NODES * HID) return;
    int row = gid / HID, col = gid - row * HID;
    float s = 1.0f + eps[layer];
    agg[gid] = s * hbase[(size_t)row * ldh + col] + agg[gid];
}

// ---------------------------------------------------------------------------
extern "C" void kernel_launch(void* const* d_in, const int* in_sizes, int n_in,
                              void* d_out, int out_size, void* d_ws, size_t ws_size,
                              hipStream_t stream) {
    (void)in_sizes; (void)n_in; (void)out_size; (void)ws_size;
    const float* x        = (const float*)d_in[0];
    const int*   eidx     = (const int*)  d_in[1];   // int32 per harness convention
    const float* eattr    = (const float*)d_in[2];
    const float* emb_w    = (const float*)d_in[3];
    const float* emb_b    = (const float*)d_in[4];
    const float* emb_g    = (const float*)d_in[5];
    const float* emb_beta = (const float*)d_in[6];
    const float* eps      = (const float*)d_in[7];
    const float* lin_e_w  = (const float*)d_in[8];
    const float* lin_e_b  = (const float*)d_in[9];
    const float* w1       = (const float*)d_in[10];
    const float* b1       = (const float*)d_in[11];
    const float* g1       = (const float*)d_in[12];
    const float* beta1    = (const float*)d_in[13];
    const float* w2       = (const float*)d_in[14];
    const float* b2       = (const float*)d_in[15];
    const float* g2       = (const float*)d_in[16];
    const float* beta2    = (const float*)d_in[17];
    float* out = (float*)d_out;

    // workspace layout: t[50000*256] | agg[50000*128] | stats[1024 floats]
    char*  ws     = (char*)d_ws;
    float* t      = (float*)ws;
    float* agg    = (float*)(ws + (size_t)N_NODES * 256 * sizeof(float));
    float* gsum   = (float*)(ws + (size_t)N_NODES * 256 * sizeof(float)
                                + (size_t)N_NODES * 128 * sizeof(float));
    float* gsumsq = gsum + 256;
    float* scale  = gsum + 512;
    float* shift  = gsum + 768;

    const int mBlocks = N_NODES / MT;   // 625 (exact)

    auto runBN = [&](const float* X, int C, const float* g, const float* b,
                     float* Y, int ldout) {
        hipMemsetAsync(gsum, 0, 2 * 256 * sizeof(float), stream);
        bn_stats<<<dim3((N_NODES + 255) / 256), 256, 0, stream>>>(X, N_NODES, C, gsum, gsumsq);
        bn_finalize<<<1, C, 0, stream>>>(gsum, gsumsq, g, b, N_NODES, scale, shift);
        int total = N_NODES * C;
        bn_apply_gelu<<<dim3((total + 255) / 256), 256, 0, stream>>>(
            X, C, Y, ldout, scale, shift, N_NODES, C);
    };

    // ---- node embed: h0 = gelu(bn(x @ emb_w + emb_b)) -> out[:, 0:128]
    gemm_wmma_f32<HID, NODE_F><<<dim3(mBlocks, HID / 64), 128, 0, stream>>>(
        x, NODE_F, emb_w, emb_b, agg, HID);
    runBN(agg, HID, emb_g, emb_beta, out + 0, OUT_C);

    // ---- GINE layers
    for (int l = 0; l < N_LAYERS; ++l) {
        const float* h = out + (size_t)l * HID;     // row stride OUT_C

        hipMemsetAsync(agg, 0, (size_t)N_NODES * HID * sizeof(float), stream);
        gine_edge_scatter<<<2048, 256, 0, stream>>>(
            eattr, eidx,
            lin_e_w + (size_t)l * EDGE_F * HID,
            lin_e_b + (size_t)l * HID,
            h, OUT_C, agg);

        int tot = N_NODES * HID;
        z0_combine<<<(tot + 255) / 256, 256, 0, stream>>>(agg, h, OUT_C, eps, l);

        // z = gelu(bn(z0 @ w1 + b1))   [N, 256], in place on t
        gemm_wmma_f32<2 * HID, HID><<<dim3(mBlocks, (2 * HID) / 64), 128, 0, stream>>>(
            agg, HID, w1 + (size_t)l * HID * 2 * HID, b1 + (size_t)l * 2 * HID,
            t, 2 * HID);
        runBN(t, 2 * HID, g1 + (size_t)l * 2 * HID, beta1 + (size_t)l * 2 * HID, t, 2 * HID);

        // h = gelu(bn(z @ w2 + b2))    [N, 128] -> out[:, (l+1)*128 : ...]
        gemm_wmma_f32<HID, 2 * HID><<<dim3(mBlocks, HID / 64), 128, 0, stream>>>(
            t, 2 * HID, w2 + (size_t)l * 2 * HID * HID, b2 + (size_t)l * HID,
            agg, HID);
        runBN(agg, HID, g2 + (size_t)l * HID, beta2 + (size_t)l * HID,
              out + (size_t)(l + 1) * HID, OUT_C);
    }
}